// GCN_47502338294086
// MI455X (gfx1250) — compile-verified
//
#include <hip/hip_runtime.h>

// Problem constants (match reference).
#define N_ENT   50000
#define DIM     128
#define NREL    11
#define NBASIS  4
#define NEDGE   1000000
#define NLAYER  2

typedef __attribute__((ext_vector_type(2))) float v2f;
typedef __attribute__((ext_vector_type(8))) float v8f;

// ---------------------------------------------------------------- zero acc
__global__ void zero_kernel(float4* __restrict__ p, int n4) {
    int i = blockIdx.x * blockDim.x + threadIdx.x;
    if (i < n4) p[i] = make_float4(0.f, 0.f, 0.f, 0.f);
}

// ------------------------------------------------- W[o][d] = sum_b c_b * bw[l,b,o,d]
// (reference quirk: rel_trans[l] -> coeff index [l][l][b])
__global__ void build_w_kernel(const float* __restrict__ bw,
                               const float* __restrict__ bc,
                               float* __restrict__ W, int l) {
    int idx = blockIdx.x * blockDim.x + threadIdx.x;   // 0 .. DIM*DIM-1
    int o = idx >> 7, d = idx & (DIM - 1);
    float s = 0.f;
#pragma unroll
    for (int b = 0; b < NBASIS; ++b) {
        float c = bc[(l * NREL + l) * NBASIS + b];
        s += c * bw[(((size_t)l * NBASIS + b) * DIM + o) * DIM + d];
    }
    W[idx] = s;
}

// ---------------------------------------------------------------- SpMM scatter
// One wave per edge: acc[src,:] += val * emb[dst,:]  (4 floats per lane).
__global__ void scatter_kernel(const float* __restrict__ emb,
                               const float* __restrict__ eval,
                               const int* __restrict__ esrc,
                               const int* __restrict__ edst,
                               float* __restrict__ acc) {
    int wave = blockIdx.x * (blockDim.x >> 5) + (threadIdx.x >> 5);
    int lane = threadIdx.x & 31;
    if (wave >= NEDGE) return;
    int   src = esrc[wave];
    int   dst = edst[wave];
    float val = eval[wave];
    const float4 v = *(const float4*)(emb + (size_t)dst * DIM + lane * 4);
    float* p = acc + (size_t)src * DIM + lane * 4;
    atomicAdd(p + 0, v.x * val);
    atomicAdd(p + 1, v.y * val);
    atomicAdd(p + 2, v.z * val);
    atomicAdd(p + 3, v.w * val);
}

// ---------------------------------------------------------------- GEMM + relu
// out[n,o] = relu( sum_d A[n,d] * W[o,d] ), via V_WMMA_F32_16X16X4_F32.
// One wave computes one 16x16 tile; K stepped by 4 -> 32 WMMA ops per wave.
__global__ void gemm_relu_kernel(const float* __restrict__ A,   // [N_ENT, DIM]
                                 const float* __restrict__ W,   // [DIM, DIM] row-major (o, d)
                                 float* __restrict__ out) {     // [N_ENT, DIM]
    int rowBase = blockIdx.x * 16;
    int colBase = (threadIdx.x >> 5) * 16;     // 8 waves -> 8 col tiles (DIM/16)
    int lane = threadIdx.x & 31;
    int r  = lane & 15;                        // A: tile row   | B: tile col j
    int kh = lane >> 4;                        // which K half (0 -> k,k+1 ; 1 -> k+2,k+3)

    const float* ap = A + (size_t)(rowBase + r) * DIM + 2 * kh;
    const float* bp = W + (size_t)(colBase + r) * DIM + 2 * kh;  // B[k][j] = W[colBase+j][k]

    v8f c = {};
#pragma unroll
    for (int k = 0; k < DIM; k += 4) {
        v2f a = *(const v2f*)(ap + k);
        v2f b = *(const v2f*)(bp + k);
        c = __builtin_amdgcn_wmma_f32_16x16x4_f32(
                /*neg_a=*/false, a, /*neg_b=*/false, b,
                /*c_mod=*/(short)0, c, /*reuse_a=*/false, /*reuse_b=*/false);
    }

    // C/D layout: VGPR v holds M = v (lanes 0-15) / M = v+8 (lanes 16-31), N = lane&15.
    float* op = out + (size_t)(rowBase + 8 * kh) * DIM + colBase + r;
#pragma unroll
    for (int v = 0; v < 8; ++v)
        op[(size_t)v * DIM] = fmaxf(c[v], 0.f);
}

// ---------------------------------------------------------------- row L2 normalize
// One wave per row; 4 floats per lane; shuffle reduction (wave32).
__global__ void normalize_kernel(float* __restrict__ out) {
    int row  = blockIdx.x * (blockDim.x >> 5) + (threadIdx.x >> 5);
    int lane = threadIdx.x & 31;
    float4 v = *(float4*)(out + (size_t)row * DIM + lane * 4);
    float s = v.x * v.x + v.y * v.y + v.z * v.z + v.w * v.w;
#pragma unroll
    for (int off = 16; off > 0; off >>= 1)
        s += __shfl_xor(s, off, 32);
    float inv = 1.0f / fmaxf(sqrtf(s), 1e-12f);
    v.x *= inv; v.y *= inv; v.z *= inv; v.w *= inv;
    *(float4*)(out + (size_t)row * DIM + lane * 4) = v;
}

// ---------------------------------------------------------------- launch
extern "C" void kernel_launch(void* const* d_in, const int* in_sizes, int n_in,
                              void* d_out, int out_size, void* d_ws, size_t ws_size,
                              hipStream_t stream) {
    const float* ent_emb = (const float*)d_in[0];   // [N, D]
    const float* bw      = (const float*)d_in[1];   // [L, B, D, D]
    const float* bc      = (const float*)d_in[2];   // [L, R, B]
    const float* eval    = (const float*)d_in[3];   // [E]
    // d_in[4] = edge_rel: mathematically dead (relation axis collapses; see analysis)
    const int*   esrc    = (const int*)d_in[5];     // [E]
    const int*   edst    = (const int*)d_in[6];     // [E]

    float* out = (float*)d_out;                     // [N, D]; also inter-layer buffer
    float* acc = (float*)d_ws;                      // [N, D] scratch
    float* W   = acc + (size_t)N_ENT * DIM;         // [D, D] scratch

    for (int l = 0; l < NLAYER; ++l) {
        zero_kernel<<<(N_ENT * DIM / 4 + 255) / 256, 256, 0, stream>>>(
            (float4*)acc, N_ENT * DIM / 4);
        build_w_kernel<<<(DIM * DIM) / 256, 256, 0, stream>>>(bw, bc, W, l);
        const float* emb_in = (l == 0) ? ent_emb : out;
        scatter_kernel<<<NEDGE / 8, 256, 0, stream>>>(emb_in, eval, esrc, edst, acc);
        gemm_relu_kernel<<<N_ENT / 16, 256, 0, stream>>>(acc, W, out);
    }
    normalize_kernel<<<N_ENT / 8, 256, 0, stream>>>(out);
}